// ChamferCUDA2_60241211293822
// MI455X (gfx1250) — compile-verified
//
#include <hip/hip_runtime.h>
#include <hip/hip_bf16.h>
#include <math.h>

// Chamfer distance on gfx1250 using V_WMMA_F32_16X16X4_F32.
//
// d2 = |p|^2 + |q|^2 - 2<p,q>.  K=4 packing:
//   A (16x4, per p1 row n):  [-2x, -2y, -2z,   1   ]
//   B (4x16, per p2 col m):  [ x',  y',  z', |q|^2 ]
// => WMMA D = |q|^2 - 2<p,q>  (everything that varies with m).
// The row-constant |p|^2 and the max(.,0) clamp commute with min_m, so both
// are applied once after the M loop -> inner loop is load + wmma + min only.

typedef __attribute__((ext_vector_type(2))) float v2f;
typedef __attribute__((ext_vector_type(8))) float v8f;

// rowmin_out[b][n] = min_m d2(P[b,n], Q[b,m]); one 16-row strip per wave.
__global__ __launch_bounds__(256) void chamfer_rowmin_kernel(
    const float* __restrict__ P,
    const float* __restrict__ Q,
    float* __restrict__ rowmin_out,
    int N, int M)
{
    const int lane   = threadIdx.x & 31;          // wave32 lane
    const int wave   = (blockIdx.x * (blockDim.x >> 5)) + (threadIdx.x >> 5);
    const int strips = N >> 4;                    // 16-row strips per batch
    const int b      = wave / strips;
    const int n0     = (wave - b * strips) << 4;

    const int  col = lane & 15;                   // row/col index within 16-group
    const bool hi  = lane >= 16;                  // K-half selector (K=2,3 vs K=0,1)

    // ---- A operand (held for the whole M loop) -------------------------------
    // 32-bit A 16x4 layout: lanes 0-15 hold (K0,K1) of row M=lane,
    //                       lanes 16-31 hold (K2,K3) of row M=lane-16.
    const float* prow = P + ((size_t)b * N + (size_t)(n0 + col)) * 3;
    const float ax = prow[0], ay = prow[1], az = prow[2];
    const float asq = ax * ax + ay * ay + az * az; // |p|^2 of row (lane&15)
    v2f A;
    A.x = hi ? (-2.0f * az) : (-2.0f * ax);
    A.y = hi ? 1.0f         : (-2.0f * ay);

    float rmin[8];
#pragma unroll
    for (int g = 0; g < 8; ++g) rmin[g] = 3.402823466e38f;

    const float* qbase = Q + (size_t)b * M * 3;

#pragma unroll 4
    for (int m0 = 0; m0 < M; m0 += 16) {
        // B operand (4x16): lanes 0-15 hold (K0,K1) of col N=lane,
        //                   lanes 16-31 hold (K2,K3) of col N=lane-16.
        const float* qrow = qbase + (size_t)(m0 + col) * 3;
        const float qx = qrow[0], qy = qrow[1], qz = qrow[2];
        const float qsq = qx * qx + qy * qy + qz * qz;
        v2f Bm;
        Bm.x = hi ? qz  : qx;
        Bm.y = hi ? qsq : qy;

        v8f acc = {}; // C = 0
        // D = |q|^2 - 2<p,q> for a 16x16 tile
        acc = __builtin_amdgcn_wmma_f32_16x16x4_f32(
            /*neg_a=*/false, A, /*neg_b=*/false, Bm,
            /*c_mod=*/(short)0, acc, /*reuse_a=*/false, /*reuse_b=*/false);

        // D layout: VGPR g = row (g + 8*hi), column = lane&15.
#pragma unroll
        for (int g = 0; g < 8; ++g)
            rmin[g] = fminf(rmin[g], acc[g]);
    }

    // ---- Finish row mins across the 16 lanes sharing each row ----------------
    // Then add the row-constant |p|^2 (held in lane == row index) and clamp.
#pragma unroll
    for (int g = 0; g < 8; ++g) {
        float v = rmin[g];
        v = fminf(v, __shfl_xor(v, 1, 32));
        v = fminf(v, __shfl_xor(v, 2, 32));
        v = fminf(v, __shfl_xor(v, 4, 32));
        v = fminf(v, __shfl_xor(v, 8, 32));
        // lane 0 needs |p|^2 of row n0+g (in lane g);
        // lane 16 needs |p|^2 of row n0+8+g (in lane 8+g).
        const float aq = __shfl(asq, (hi ? 8 : 0) + g, 32);
        rmin[g] = fmaxf(v + aq, 0.0f);
    }
    // lane 0 holds rows n0..n0+7 (VGPR g -> row g); lane 16 holds rows n0+8..n0+15.
    if (col == 0) {
        const int rowbase = n0 + (hi ? 8 : 0);
        float* out = rowmin_out + (size_t)b * N + rowbase;
#pragma unroll
        for (int g = 0; g < 8; ++g) out[g] = rmin[g];
    }
}

// out[0] = s1 * sum(d1[0..n1)) + s2 * sum(d2[0..n2))
__global__ __launch_bounds__(256) void chamfer_reduce_kernel(
    const float* __restrict__ d1, int n1, float s1,
    const float* __restrict__ d2, int n2, float s2,
    float* __restrict__ out)
{
    __shared__ float smem[256];
    float a1 = 0.0f, a2 = 0.0f;
    for (int i = threadIdx.x; i < n1; i += 256) a1 += d1[i];
    for (int i = threadIdx.x; i < n2; i += 256) a2 += d2[i];
    smem[threadIdx.x] = s1 * a1 + s2 * a2;
    __syncthreads();
    for (int off = 128; off > 0; off >>= 1) {
        if (threadIdx.x < off) smem[threadIdx.x] += smem[threadIdx.x + off];
        __syncthreads();
    }
    if (threadIdx.x == 0) out[0] = smem[0];
}

extern "C" void kernel_launch(void* const* d_in, const int* in_sizes, int n_in,
                              void* d_out, int out_size, void* d_ws, size_t ws_size,
                              hipStream_t stream) {
    const float* p1 = (const float*)d_in[0]; // [B, N, 3]
    const float* p2 = (const float*)d_in[1]; // [B, M, 3]

    const int B = 16;
    const int N = in_sizes[0] / (B * 3);     // 4096
    const int M = in_sizes[1] / (B * 3);     // 4096

    float* dist1 = (float*)d_ws;             // [B, N]
    float* dist2 = dist1 + (size_t)B * N;    // [B, M]

    // Pass 1: dist1[b][n] = min_m d2   (16 rows per wave, 8 waves per block)
    {
        const int waves  = B * (N / 16);
        const int blocks = waves / 8;
        chamfer_rowmin_kernel<<<blocks, 256, 0, stream>>>(p1, p2, dist1, N, M);
    }
    // Pass 2 (roles swapped): dist2[b][m] = min_n d2
    {
        const int waves  = B * (M / 16);
        const int blocks = waves / 8;
        chamfer_rowmin_kernel<<<blocks, 256, 0, stream>>>(p2, p1, dist2, M, N);
    }
    // Final scalar: sum_b [ mean_n dist1 + mean_m dist2 ]
    chamfer_reduce_kernel<<<1, 256, 0, stream>>>(
        dist1, B * N, 1.0f / (float)N,
        dist2, B * M, 1.0f / (float)M,
        (float*)d_out);
}